// CSTR_40991167873338
// MI455X (gfx1250) — compile-verified
//
#include <hip/hip_runtime.h>
#include <cstdint>

// ---------------------------------------------------------------------------
// Gated-control RK4 trajectory cost, MI455X (gfx1250).
//
// Strategy: one lane per trajectory (B=8192 -> 256 waves). Since occupancy is
// ~1 wave/SIMD, memory latency cannot be hidden by TLP; instead the gfx1250
// async global->LDS DMA engine streams each lane's disturbance data one chunk
// (32 timesteps) ahead of the sequential RK4 recurrence, double-buffered in
// LDS, synchronized only with s_wait_asynccnt.
// ---------------------------------------------------------------------------

#define T_STEPS   2048
#define BLOCKT    64                    // threads per block (2 waves)
#define CHUNK     32                    // timesteps staged per chunk
#define NCHUNK    (T_STEPS / CHUNK)     // 64
#define CHUNKLETS (CHUNK / 4)           // 8 float4 groups per chunk per row
// LDS floats per buffer: [chunklet][lane][row][4] = 8 * 64 * 2 * 4
#define BUF_FLOATS (CHUNKLETS * BLOCKT * 8)   // 4096 floats = 16 KB

#define H_STEP 0.01f
#define LOG2E  1.4426950408889634f

__global__ __launch_bounds__(BLOCKT)
void CSTR_traj_kernel(const float* __restrict__ w,
                      const float* __restrict__ K,
                      const float* __restrict__ L,
                      const float* __restrict__ M,
                      const float* __restrict__ Mo,
                      float* __restrict__ out, int B)
{
    __shared__ alignas(16) float smem[2 * BUF_FLOATS];   // 32 KB double buffer

    const int tid = (int)threadIdx.x;
    const int bRaw = (int)blockIdx.x * BLOCKT + tid;
    const int b = bRaw < B ? bRaw : (B - 1);             // clamp (keeps EXEC full)

    // --- tiny shared coefficients into registers (uniform -> scalar loads) ---
    const float K0 = K[0], K1 = K[1];
    float Lc[16];
#pragma unroll
    for (int i = 0; i < 16; ++i) Lc[i] = L[i];
    const float M0v = M[0], M1v = M[1], M2v = M[2], M3v = M[3];
    const float MoV = Mo[0];

    // w layout: (B, 2, T). Row0 base for this lane; row1 at +T floats.
    const float* wbase = w + (size_t)b * 2u * T_STEPS;
    const uint32_t lds0 = (uint32_t)(uintptr_t)smem;     // low 32 bits == LDS byte offset

    // Issue one chunk of async global->LDS copies (16 x b128 per lane).
    // LDS byte offset for (cc,row): buf*16KB + cc*(BLOCKT*32) + tid*32 + row*16.
    auto issue = [&](int c, int buf) {
        const float* g0 = wbase + c * CHUNK;
        const uint32_t lbase = lds0 + (uint32_t)buf * (BUF_FLOATS * 4u)
                                    + (uint32_t)tid * 32u;
#pragma unroll
        for (int cc = 0; cc < CHUNKLETS; ++cc) {
            const uint64_t ga0 = (uint64_t)(uintptr_t)(g0 + cc * 4);
            const uint64_t ga1 = (uint64_t)(uintptr_t)(g0 + T_STEPS + cc * 4);
            const uint32_t la  = lbase + (uint32_t)cc * (BLOCKT * 32u);
            asm volatile("global_load_async_to_lds_b128 %0, %1, off"
                         :: "v"(la), "v"(ga0) : "memory");
            asm volatile("global_load_async_to_lds_b128 %0, %1, off"
                         :: "v"(la + 16u), "v"(ga1) : "memory");
        }
    };

    issue(0, 0);                                         // prime the pipeline

    // State: x = (1,0), x_hat = x0.
    float x1 = 1.0f, x2 = 0.0f, xh1 = 1.0f, xh2 = 0.0f;
    float J = 0.0f, sc = 0.0f;

    for (int c = 0; c < NCHUNK; ++c) {
        const int buf = c & 1;
        if (c + 1 < NCHUNK) {
            // Buffer being refilled was last read at chunk c-1; drain DS queue.
            asm volatile("s_wait_dscnt 0" ::: "memory");
            issue(c + 1, buf ^ 1);
            // 32 outstanding (c and c+1); async loads retire in order, so
            // waiting to <=16 guarantees chunk c has landed in LDS.
            asm volatile("s_wait_asynccnt 16" ::: "memory");
        } else {
            asm volatile("s_wait_asynccnt 0" ::: "memory");
        }

        const float4* lp = (const float4*)(smem + buf * BUF_FLOATS + tid * 8);
#pragma unroll
        for (int cc = 0; cc < CHUNKLETS; ++cc) {
            const float4 w0 = lp[cc * (BLOCKT * 2)];         // row 0, 4 steps
            const float4 w1 = lp[cc * (BLOCKT * 2) + 1];     // row 1, 4 steps
#pragma unroll
            for (int j = 0; j < 4; ++j) {
                const int gt = c * CHUNK + cc * 4 + j;
                const float wv1 = (&w0.x)[j];
                const float wv2 = (&w1.x)[j];

                // ---- gate: phi = z^T L z + M z + Mo,  z = [x1,x2,xh1,xh2] ----
                float r0 = fmaf(Lc[0],  x1, fmaf(Lc[1],  x2, fmaf(Lc[2],  xh1, Lc[3]  * xh2)));
                float r1 = fmaf(Lc[4],  x1, fmaf(Lc[5],  x2, fmaf(Lc[6],  xh1, Lc[7]  * xh2)));
                float r2 = fmaf(Lc[8],  x1, fmaf(Lc[9],  x2, fmaf(Lc[10], xh1, Lc[11] * xh2)));
                float r3 = fmaf(Lc[12], x1, fmaf(Lc[13], x2, fmaf(Lc[14], xh1, Lc[15] * xh2)));
                float phi = fmaf(r0, x1, fmaf(r1, x2, fmaf(r2, xh1, r3 * xh2)));
                phi = fmaf(M0v, x1, phi);
                phi = fmaf(M1v, x2, phi);
                phi = fmaf(M2v, xh1, phi);
                phi = fmaf(M3v, xh2, phi);
                phi += MoV;
                // sigmoid via v_exp_f32 / v_rcp_f32 transcendentals
                const float sig =
                    __builtin_amdgcn_rcpf(1.0f + __builtin_amdgcn_exp2f(-LOG2E * phi));
                // t==0: delta := 1 and x_hat unchanged. Since x==x_hat at t==0,
                // the gated update below is the identity there; only the cost
                // term needs delta forced to 1.
                const float delta = (gt == 0) ? 1.0f : sig;
                xh1 = fmaf(delta, x1 - xh1, xh1);
                xh2 = fmaf(delta, x2 - xh2, xh2);

                // ---- control from gated estimate ----
                const float u = fmaf(K1, xh2, K0 * xh1);

                // ---- RK4 (faithful to reference: k3 == k2) ----
                const float k11 = fmaf(-2.0f, x1, x2) + u;
                const float k12 = fmaf(-2.0f, x2, x1) + u;
                const float x2b = x2 + (H_STEP * 0.5f);
                const float k21 = fmaf(-2.0f, x1, x2b) + u;
                const float k22 = fmaf(-2.0f, x2b, x1) + u;
                const float x2c = x2 + H_STEP;
                const float k41 = fmaf(-2.0f, x1, x2c) + u;
                const float k42 = fmaf(-2.0f, x2c, x1) + u;
                const float s1 = fmaf(4.0f, k21, k11 + k41);   // k1+4k2+k4
                const float s2 = fmaf(4.0f, k22, k12 + k42);

                // ---- stage cost on pre-update state ----
                const float us = fmaf(K1, x2, K0 * x1);
                sc = x1 * x1;
                sc = fmaf(x2, x2, sc);
                sc = fmaf(0.1f * us, us, sc);
                sc += delta;                                   // LAM = 1
                J += sc;

                // ---- state update + disturbance ----
                x1 = fmaf(H_STEP / 6.0f, s1, x1) + wv1;
                x2 = fmaf(H_STEP / 6.0f, s2, x2) + wv2;
            }
        }
    }

    J -= sc;                                   // reference sums scs[:T-1]
    J = fmaf(10.0f, fmaf(x1, x1, x2 * x2), J); // terminal x_T^T Qf x_T

    if (bRaw < B) out[bRaw] = J;
}

extern "C" void kernel_launch(void* const* d_in, const int* in_sizes, int n_in,
                              void* d_out, int out_size, void* d_ws, size_t ws_size,
                              hipStream_t stream) {
    const float* w  = (const float*)d_in[0];   // (B, 2, 2048)
    const float* K  = (const float*)d_in[1];   // (1, 2)
    const float* L  = (const float*)d_in[2];   // (4, 4)
    const float* M  = (const float*)d_in[3];   // (1, 4)
    const float* Mo = (const float*)d_in[4];   // (1, 1)
    float* out = (float*)d_out;

    const int B = in_sizes[0] / (2 * T_STEPS);
    const int grid = (B + BLOCKT - 1) / BLOCKT;
    CSTR_traj_kernel<<<grid, BLOCKT, 0, stream>>>(w, K, L, M, Mo, out, B);
}